// MultiBoxLoss_22986664968667
// MI455X (gfx1250) — compile-verified
//
#include <hip/hip_runtime.h>
#include <hip/hip_bf16.h>
#include <stdint.h>

#define DD 8732          // priors
#define BB 128           // batch
#define MM 32            // gt boxes
#define CC 21            // classes
#define NBLK 256         // threads per block (8 waves of 32)
#define NWAVES 8
#define TAIL0 8704       // (DD/NBLK)*NBLK
#define NCHUNK 34        // TAIL0/NBLK

#if defined(__HIP_DEVICE_COMPILE__) && __has_builtin(__builtin_amdgcn_global_load_async_to_lds_b128)
#define HAS_ASYNC 1
#else
#define HAS_ASYNC 0
#endif

// ---------------- wave reduction: WMMA ones-matrix column-sum + shfl ----------
// D = ones(16x4) x B + 0  ->  D[i][j] = sum_k B[k][j]; all rows of D identical.
// Sum of D-row-0 over lanes 0..15 == sum of ALL 64 B slots, independent of the
// exact B slot<->(k,j) mapping -> exact 32-lane sum.
__device__ __forceinline__ float waveRed32(float v) {
#if defined(__HIP_DEVICE_COMPILE__) && __has_builtin(__builtin_amdgcn_wmma_f32_16x16x4_f32)
  typedef float v2f __attribute__((ext_vector_type(2)));
  typedef float v8f __attribute__((ext_vector_type(8)));
  v2f a; a[0] = 1.0f; a[1] = 1.0f;      // A = 16x4 all-ones
  v2f bm; bm[0] = v; bm[1] = 0.0f;      // B = lane partials (rest zero)
  v8f c = {};
  v8f dacc = __builtin_amdgcn_wmma_f32_16x16x4_f32(
      false, a, false, bm, (short)0, c, false, false);
  float s = dacc[0];
  s += __shfl_xor(s, 8, 32);
  s += __shfl_xor(s, 4, 32);
  s += __shfl_xor(s, 2, 32);
  s += __shfl_xor(s, 1, 32);
  return s;                              // lane 0 holds the full 32-lane sum
#else
  for (int m = 16; m; m >>= 1) v += __shfl_xor(v, m, 32);
  return v;
#endif
}

// ---- issue 6 x b128 async DMA loads (3072B tile of cls_pred -> LDS), no wait ----
__device__ __forceinline__ void stageIssue(const float* gsrc, float* lbuf, int lane) {
#if HAS_ASYNC
  typedef int v4i_t __attribute__((vector_size(16)));           // matches builtin pointee
  typedef __attribute__((address_space(1))) v4i_t gv4i_t;       // global (AS1)
  typedef __attribute__((address_space(3))) v4i_t lv4i_t;       // LDS (AS3)
  gv4i_t* g = (gv4i_t*)(unsigned long long)(uintptr_t)gsrc;
  lv4i_t* l = (lv4i_t*)(unsigned)(uintptr_t)lbuf;               // flat->LDS: low 32 bits
#pragma unroll
  for (int j = 0; j < 6; ++j) {                                 // uniform: exactly 6 issues
    int i = lane + j * 32;
    __builtin_amdgcn_global_load_async_to_lds_b128(g + i, l + i, 0, 0);
  }
#else
  (void)gsrc; (void)lbuf; (void)lane;
#endif
}

__device__ __forceinline__ void waitAsyncAllButNext() {         // keep newest 6 in flight
#if HAS_ASYNC
#if __has_builtin(__builtin_amdgcn_s_wait_asynccnt)
  __builtin_amdgcn_s_wait_asynccnt(6);
#else
  asm volatile("s_wait_asynccnt 6" ::: "memory");
#endif
  asm volatile("" ::: "memory");
#endif
}

__device__ __forceinline__ void waitAsyncAll() {
#if HAS_ASYNC
#if __has_builtin(__builtin_amdgcn_s_wait_asynccnt)
  __builtin_amdgcn_s_wait_asynccnt(0);
#else
  asm volatile("s_wait_asynccnt 0" ::: "memory");
#endif
  asm volatile("" ::: "memory");
#endif
}

__device__ __forceinline__ float sl1(float x) {
  float a = fabsf(x);
  return (a < 1.0f) ? (0.5f * x * x) : (a - 0.5f);
}

__global__ __launch_bounds__(NBLK, 1)
void mbx_main(const float* __restrict__ locp, const float* __restrict__ cls,
              const float* __restrict__ gtb, const int* __restrict__ gtl,
              const float* __restrict__ dflt,
              float* __restrict__ blkLoss, int* __restrict__ blkNp) {
  const int b = blockIdx.x;
  const int tid = (int)threadIdx.x;
  const int lane = tid & 31;
  const int wv = tid >> 5;

  __shared__ unsigned sCe[DD];                 // phase1: packed (iou,m); phase2: CE bits
  __shared__ float4  sStage[NWAVES * 384];     // per-wave double buffer: 2 x 192 x b128
  __shared__ float4  sBox[MM];
  __shared__ float   sArea[MM];
  __shared__ int     sLab[MM];
  __shared__ unsigned long long sMKey[MM];
  __shared__ int     sDpg[MM];
  __shared__ int     sHist[256];
  __shared__ float   sRed[NBLK];
  __shared__ unsigned sPrefix;
  __shared__ int     sRem;
  __shared__ float   sLocS, sCpS, sConfNeg;
  __shared__ int     sNp, sNneg;

  if (tid < MM) {
    float4 g = ((const float4*)(gtb + (size_t)b * MM * 4))[tid];
    sBox[tid] = g;
    sArea[tid] = (g.z - g.x) * (g.w - g.y);
    sLab[tid] = gtl[(size_t)b * MM + tid];
    sMKey[tid] = 0ULL;
  }
  __syncthreads();

  // ---------------- Phase 1: IoU matching --------------------------------
  unsigned long long bk[MM];
#pragma unroll
  for (int m = 0; m < MM; ++m) bk[m] = 0ULL;

  for (int d = tid; d < DD; d += NBLK) {
    float4 pr = ((const float4*)dflt)[d];
    float hx = pr.z * 0.5f, hy = pr.w * 0.5f;
    float px0 = pr.x - hx, py0 = pr.y - hy;
    float px1 = pr.x + hx, py1 = pr.y + hy;
    float pA = pr.z * pr.w;
    float bi = -1.0f; int bm = 0;
#pragma unroll
    for (int m = 0; m < MM; ++m) {
      float4 g = sBox[m];
      float ltx = fmaxf(g.x, px0), lty = fmaxf(g.y, py0);
      float rbx = fminf(g.z, px1), rby = fminf(g.w, py1);
      float iw = fmaxf(rbx - ltx, 0.0f), ih = fmaxf(rby - lty, 0.0f);
      float inter = iw * ih;
      float iou = inter * __builtin_amdgcn_rcpf(sArea[m] + pA - inter);
      if (iou > bi) { bi = iou; bm = m; }                    // first-index argmax
      unsigned long long k =
          ((unsigned long long)__float_as_uint(iou) << 32) | (unsigned)(DD - 1 - d);
      if (k > bk[m]) bk[m] = k;                              // per-gt best prior
    }
    sCe[d] = (__float_as_uint(bi) & 0xFFFFFFE0u) | (unsigned)bm;  // pack iou(hi)+m(lo5)
  }
#pragma unroll
  for (int m = 0; m < MM; ++m) atomicMax(&sMKey[m], bk[m]);
  __syncthreads();
  if (tid < MM) sDpg[tid] = (DD - 1) - (int)(unsigned)(sMKey[tid] & 0xFFFFFFFFu);
  __syncthreads();

  // ---------------- Phase 2: CE + loc loss (streams cls_pred once) --------
  float locAcc = 0.0f, cpsAcc = 0.0f, nposAcc = 0.0f;

  auto processRow = [&](int d, const float* rp) {
    unsigned pk = sCe[d];
    int m = (int)(pk & 31u);
    float iou = __uint_as_float(pk & 0xFFFFFFE0u);
#pragma unroll
    for (int j = 0; j < MM; ++j)
      if (sDpg[j] == d) { m = j; iou = 1.0f; }               // force-match (last wins)
    int lab = (iou < 0.5f) ? 0 : sLab[m];
    float mx = rp[0];
#pragma unroll
    for (int k = 1; k < CC; ++k) mx = fmaxf(mx, rp[k]);
    float se = 0.0f;
#pragma unroll
    for (int k = 0; k < CC; ++k) se += __expf(rp[k] - mx);
    float ce = fmaxf(mx + __logf(se) - rp[lab], 0.0f);
    if (lab > 0) {
      cpsAcc += ce; nposAcc += 1.0f;
      float4 g = sBox[m];
      float4 pr = ((const float4*)dflt)[d];
      float rw = __builtin_amdgcn_rcpf(pr.z), rh = __builtin_amdgcn_rcpf(pr.w);
      float t0 = (((g.x + g.z) * 0.5f) - pr.x) * 10.0f * rw;
      float t1 = (((g.y + g.w) * 0.5f) - pr.y) * 10.0f * rh;
      float t2 = __logf((g.z - g.x) * rw) * 5.0f;
      float t3 = __logf((g.w - g.y) * rh) * 5.0f;
      float4 lp = ((const float4*)locp)[(size_t)b * DD + d];
      locAcc += sl1(lp.x - t0) + sl1(lp.y - t1) + sl1(lp.z - t2) + sl1(lp.w - t3);
      sCe[d] = 0u;                                           // positives: ce_neg = 0
    } else {
      sCe[d] = __float_as_uint(ce);
    }
  };

#if HAS_ASYNC
  // 2-deep async pipeline: prefetch chunk c+1, wait asynccnt<=6 (in-order
  // completion => chunk c landed), process chunk c while c+1 streams in.
  float* bufA = (float*)(sStage + (size_t)wv * 384);
  float* bufB = bufA + 192 * 4;
  stageIssue(cls + ((size_t)b * DD + (wv * 32)) * CC, bufA, lane);
  for (int c = 0; c < NCHUNK; ++c) {
    int d0 = c * NBLK + wv * 32;
    float* cur = (c & 1) ? bufB : bufA;
    if (c + 1 < NCHUNK) {
      int d1 = d0 + NBLK;
      stageIssue(cls + ((size_t)b * DD + d1) * CC, (c & 1) ? bufA : bufB, lane);
      waitAsyncAllButNext();
    } else {
      waitAsyncAll();
    }
    processRow(d0 + lane, cur + lane * CC);
  }
#else
  float* lbuf = (float*)(sStage + (size_t)wv * 384);
  for (int c = 0; c < NCHUNK; ++c) {
    int d0 = c * NBLK + wv * 32;
    const float* gsrc = cls + ((size_t)b * DD + d0) * CC;
    for (int i = lane; i < 168; i += 32)
      ((float4*)lbuf)[i] = ((const float4*)gsrc)[i];
    __syncthreads();  // uniform across the block
    processRow(d0 + lane, lbuf + lane * CC);
  }
#endif
  if (tid < DD - TAIL0) {                                    // 28 tail rows, direct loads
    int d = TAIL0 + tid;
    processRow(d, cls + ((size_t)b * DD + d) * CC);
  }
  __syncthreads();

  // ---------------- block reductions (WMMA path) --------------------------
  auto blockRed = [&](float v) -> float {
    sRed[tid] = v;
    __syncthreads();
    float s = 0.0f;
    if (tid < 32) {
#pragma unroll
      for (int i = 0; i < 8; ++i) s += sRed[tid + 32 * i];
      s = waveRed32(s);
    }
    __syncthreads();
    return s;                                                // valid on tid==0
  };

  float locT = blockRed(locAcc);
  float cpT  = blockRed(cpsAcc);
  float npT  = blockRed(nposAcc);
  if (tid == 0) {
    sLocS = locT; sCpS = cpT;
    int np = (int)(npT + 0.5f);
    sNp = np; sNneg = 3 * np; sConfNeg = 0.0f;
  }
  __syncthreads();

  // ---------------- hard-negative mining: radix-select top n_neg ----------
  int nneg = sNneg;
  if (nneg > 0) {
    if (nneg >= DD) {
      float p = 0.0f;
      for (int d = tid; d < DD; d += NBLK) p += __uint_as_float(sCe[d]);
      float t = blockRed(p);
      if (tid == 0) sConfNeg = t;
    } else {
      unsigned prefix = 0; int r = nneg;
      for (int pass = 3; pass >= 0; --pass) {
        int shift = pass * 8;
        unsigned hiMask = (pass == 3) ? 0u : (0xFFFFFFFFu << (shift + 8));
        sHist[tid] = 0;
        __syncthreads();
        for (int d = tid; d < DD; d += NBLK) {
          unsigned k = sCe[d];
          if ((k & hiMask) == (prefix & hiMask))
            atomicAdd(&sHist[(k >> shift) & 255], 1);
        }
        __syncthreads();
        if (tid == 0) {
          int cum = 0, bin = 0;
          for (int q = 255; q >= 0; --q) {
            int h = sHist[q];
            if (cum + h >= r) { bin = q; break; }
            cum += h;
          }
          sPrefix = prefix | ((unsigned)bin << shift);
          sRem = r - cum;
        }
        __syncthreads();
        prefix = sPrefix; r = sRem;
      }
      // sum strictly greater than threshold; ties contribute r * value (exact)
      float p = 0.0f;
      for (int d = tid; d < DD; d += NBLK) {
        unsigned k = sCe[d];
        if (k > prefix) p += __uint_as_float(k);
      }
      float t = blockRed(p);
      if (tid == 0) sConfNeg = t + (float)r * __uint_as_float(prefix);
    }
  }
  __syncthreads();
  if (tid == 0) {
    blkLoss[b] = sLocS + sCpS + sConfNeg;
    blkNp[b] = sNp;
  }
}

__global__ void mbx_final(const float* __restrict__ blkLoss,
                          const int* __restrict__ blkNp,
                          float* __restrict__ out) {
  if (threadIdx.x == 0 && blockIdx.x == 0) {
    float s = 0.0f; long long np = 0;
    for (int i = 0; i < BB; ++i) { s += blkLoss[i]; np += blkNp[i]; }  // fixed order
    if (np < 1) np = 1;
    out[0] = s / (float)np;
  }
}

extern "C" void kernel_launch(void* const* d_in, const int* in_sizes, int n_in,
                              void* d_out, int out_size, void* d_ws, size_t ws_size,
                              hipStream_t stream) {
  (void)in_sizes; (void)n_in; (void)out_size; (void)ws_size;
  const float* loc_pred = (const float*)d_in[0];
  const float* cls_pred = (const float*)d_in[1];
  const float* gt_boxes = (const float*)d_in[2];
  const int*   gt_labels = (const int*)d_in[3];
  const float* default_boxes = (const float*)d_in[4];

  float* blkLoss = (float*)d_ws;
  int*   blkNp   = (int*)((char*)d_ws + BB * sizeof(float));

  mbx_main<<<dim3(BB), dim3(NBLK), 0, stream>>>(loc_pred, cls_pred, gt_boxes,
                                                gt_labels, default_boxes,
                                                blkLoss, blkNp);
  mbx_final<<<dim3(1), dim3(32), 0, stream>>>(blkLoss, blkNp, (float*)d_out);
}